// Net_80496277062111
// MI455X (gfx1250) — compile-verified
//
#include <hip/hip_runtime.h>
#include <hip/hip_bf16.h>

#define NN 50000
#define NE 800000
#define F  128
#define NC 16
#define ROW_TILES (NN / 16)   // 3125, exact

typedef float v2f __attribute__((ext_vector_type(2)));
typedef float v8f __attribute__((ext_vector_type(8)));

// ---------------- degree / normalization ----------------
__global__ void k_set_deg_one(float* deg) {
    int i = blockIdx.x * 256 + threadIdx.x;
    if (i < NN) deg[i] = 1.0f;  // self-loop contributes 1
}

__global__ void k_count_deg(const int* __restrict__ ei, float* deg) {
    int e = blockIdx.x * 256 + threadIdx.x;
    if (e < NE) atomicAdd(&deg[ei[NE + e]], 1.0f);  // dst row
}

__global__ void k_deg_to_dinv(float* deg) {
    int i = blockIdx.x * 256 + threadIdx.x;
    if (i < NN) deg[i] = rsqrtf(deg[i]);  // deg >= 1 always
}

// ---------------- dense transform: C[N,128] = act(A) @ W, WMMA f32 16x16x4 ----------------
// IN_ACT : A element = relu(A + inBias[k])   (previous layer's bias+ReLU fused)
// OUT_ACT: C element = relu(C + outBias[n])  (MLP bias+ReLU fused)
template <bool IN_ACT, bool OUT_ACT>
__global__ __launch_bounds__(256)
void k_gemm_wmma(const float* __restrict__ A, const float* __restrict__ W,
                 float* __restrict__ C,
                 const float* __restrict__ inBias,
                 const float* __restrict__ outBias) {
    const int lane = threadIdx.x & 31;
    const int wave = threadIdx.x >> 5;          // 8 waves -> 8 column tiles of 16
    const int rowTile = blockIdx.x;             // 16 rows per block
    const int m    = rowTile * 16 + (lane & 15);
    const int koff = (lane >> 4) * 2;           // lanes 0-15: K=0,1 ; lanes 16-31: K=2,3
    const int n    = wave * 16 + (lane & 15);

    // hoisted per-lane base pointers: per k-step all offsets are compile-time constants
    const float* __restrict__ aRow = A + (size_t)m * F + koff;  // step +4 floats
    const float* __restrict__ wCol = W + (size_t)koff * F + n;  // step +4 rows = +512 floats
    const float* __restrict__ bK   = IN_ACT ? (inBias + koff) : nullptr;

    v8f acc = {};
    #pragma unroll
    for (int k0 = 0; k0 < F; k0 += 4) {
        float a0 = aRow[k0];
        float a1 = aRow[k0 + 1];
        if (IN_ACT) {
            a0 = fmaxf(a0 + bK[k0], 0.0f);
            a1 = fmaxf(a1 + bK[k0 + 1], 0.0f);
        }
        v2f av; av.x = a0; av.y = a1;
        v2f bv; bv.x = wCol[k0 * F]; bv.y = wCol[k0 * F + F];
        acc = __builtin_amdgcn_wmma_f32_16x16x4_f32(false, av, false, bv,
                                                    (short)0, acc, false, false);
    }

    const int half = lane >> 4;
    const float ob = OUT_ACT ? outBias[n] : 0.0f;
    #pragma unroll
    for (int r = 0; r < 8; ++r) {
        const int row = rowTile * 16 + r + half * 8;
        float v = acc[r];
        if (OUT_ACT) v = fmaxf(v + ob, 0.0f);
        C[(size_t)row * F + n] = v;
    }
}

// ---------------- aggregation: AGG = D^-1/2 (A+I) D^-1/2 @ H ----------------
// self-loop term initializes AGG (also deterministically overwrites it each call)
__global__ void k_self_init(const float* __restrict__ H, const float* __restrict__ dinv,
                            float* __restrict__ AGG) {
    int t = blockIdx.x * 256 + threadIdx.x;     // NN*32 threads, 4 floats each
    int i = t >> 5;
    int f = (t & 31) * 4;
    if (i < NN) {
        float s = dinv[i]; s = s * s;
        const float4 h = *(const float4*)(H + (size_t)i * F + f);
        float4 o; o.x = h.x * s; o.y = h.y * s; o.z = h.z * s; o.w = h.w * s;
        *(float4*)(AGG + (size_t)i * F + f) = o;
    }
}

__global__ void k_edge_scatter(const int* __restrict__ ei, const float* __restrict__ H,
                               const float* __restrict__ dinv, float* __restrict__ AGG) {
    int t = blockIdx.x * 256 + threadIdx.x;     // NE*32 threads, 4 floats each
    int e = t >> 5;
    int f = (t & 31) * 4;
    if (e < NE) {
        const int s = ei[e];            // src
        const int d = ei[NE + e];       // dst
        const float nrm = dinv[s] * dinv[d];
        const float4 h = *(const float4*)(H + (size_t)s * F + f);
        float* out = AGG + (size_t)d * F + f;
        atomicAdd(out + 0, h.x * nrm);
        atomicAdd(out + 1, h.y * nrm);
        atomicAdd(out + 2, h.z * nrm);
        atomicAdd(out + 3, h.w * nrm);
    }
}

// ---------------- final 128->16 + log_softmax, one 16x16 WMMA tile per wave ----------------
__global__ __launch_bounds__(256)
void k_final_wmma(const float* __restrict__ A, const float* __restrict__ Wf,
                  const float* __restrict__ bf, float* __restrict__ out) {
    const int lane = threadIdx.x & 31;
    const int wave = threadIdx.x >> 5;
    const int rowTile = blockIdx.x * 8 + wave;
    if (rowTile >= ROW_TILES) return;           // wave-uniform guard, EXEC stays full

    const int m16  = lane & 15;
    const int koff = (lane >> 4) * 2;
    const int n    = m16;                       // class index 0..15

    const float* __restrict__ aRow = A + (size_t)(rowTile * 16 + m16) * F + koff;
    const float* __restrict__ wCol = Wf + (size_t)koff * NC + n;  // step +4 rows = +64 floats

    v8f acc = {};
    #pragma unroll
    for (int k0 = 0; k0 < F; k0 += 4) {
        v2f av; av.x = aRow[k0];        av.y = aRow[k0 + 1];
        v2f bv; bv.x = wCol[k0 * NC];   bv.y = wCol[k0 * NC + NC];
        acc = __builtin_amdgcn_wmma_f32_16x16x4_f32(false, av, false, bv,
                                                    (short)0, acc, false, false);
    }

    const int half = lane >> 4;
    const float bias = bf[n];
    #pragma unroll
    for (int r = 0; r < 8; ++r) {
        float v = acc[r] + bias;
        // log_softmax over the 16 classes = 16 lanes of this half (xor masks stay in-half)
        float mx = v;
        for (int off = 1; off < 16; off <<= 1) mx = fmaxf(mx, __shfl_xor(mx, off, 32));
        float ex = __expf(v - mx);
        float sm = ex;
        for (int off = 1; off < 16; off <<= 1) sm += __shfl_xor(sm, off, 32);
        const float lse = mx + __logf(sm);
        const int row = rowTile * 16 + r + half * 8;
        out[(size_t)row * NC + n] = v - lse;
    }
}

// ---------------- driver ----------------
extern "C" void kernel_launch(void* const* d_in, const int* in_sizes, int n_in,
                              void* d_out, int out_size, void* d_ws, size_t ws_size,
                              hipStream_t stream) {
    (void)in_sizes; (void)n_in; (void)out_size; (void)ws_size;

    const float* x   = (const float*)d_in[0];
    const int*   ei  = (const int*)  d_in[1];
    // d_in[2] = TRAIN scalar (unused; reference uses deterministic path)
    const float* W1  = (const float*)d_in[3];  const float* b1  = (const float*)d_in[4];
    const float* W2  = (const float*)d_in[5];  const float* b2  = (const float*)d_in[6];
    const float* W3  = (const float*)d_in[7];  const float* b3  = (const float*)d_in[8];
    const float* Wf1 = (const float*)d_in[9];  const float* bf1 = (const float*)d_in[10];
    const float* Wf2 = (const float*)d_in[11]; const float* bf2 = (const float*)d_in[12];
    const float* Wf3 = (const float*)d_in[13]; const float* bf3 = (const float*)d_in[14];
    float* out = (float*)d_out;

    // workspace layout (floats): dinv[50176 pad] | H[NN*F] | AGG[NN*F]
    float* dinv = (float*)d_ws;
    float* H    = dinv + 50176;
    float* AGG  = H + (size_t)NN * F;

    const int nb_nodes = (NN + 255) / 256;
    const int nb_edges = (NE + 255) / 256;
    const int nb_selfi = (NN * 32 + 255) / 256;
    const int nb_scat  = (NE * 32 + 255) / 256;
    const int nb_final = (ROW_TILES + 7) / 8;

    // normalization: deg = 1 + indegree ; dinv = rsqrt(deg)
    k_set_deg_one<<<nb_nodes, 256, 0, stream>>>(dinv);
    k_count_deg  <<<nb_edges, 256, 0, stream>>>(ei, dinv);
    k_deg_to_dinv<<<nb_nodes, 256, 0, stream>>>(dinv);

    // GCN layer 1: h = x @ W1
    k_gemm_wmma<false, false><<<ROW_TILES, 256, 0, stream>>>(x, W1, H, nullptr, nullptr);
    k_self_init   <<<nb_selfi, 256, 0, stream>>>(H, dinv, AGG);
    k_edge_scatter<<<nb_scat,  256, 0, stream>>>(ei, H, dinv, AGG);
    // GCN layer 2: h = relu(agg1 + b1) @ W2   (bias+ReLU fused into GEMM input)
    k_gemm_wmma<true, false><<<ROW_TILES, 256, 0, stream>>>(AGG, W2, H, b1, nullptr);
    k_self_init   <<<nb_selfi, 256, 0, stream>>>(H, dinv, AGG);
    k_edge_scatter<<<nb_scat,  256, 0, stream>>>(ei, H, dinv, AGG);
    // GCN layer 3: h = relu(agg2 + b2) @ W3
    k_gemm_wmma<true, false><<<ROW_TILES, 256, 0, stream>>>(AGG, W3, H, b2, nullptr);
    k_self_init   <<<nb_selfi, 256, 0, stream>>>(H, dinv, AGG);
    k_edge_scatter<<<nb_scat,  256, 0, stream>>>(ei, H, dinv, AGG);

    // MLP head: y = relu(relu(agg3 + b3) @ Wf1 + bf1)
    k_gemm_wmma<true, true><<<ROW_TILES, 256, 0, stream>>>(AGG, Wf1, H, b3, bf1);
    // y = relu(y @ Wf2 + bf2)
    k_gemm_wmma<false, true><<<ROW_TILES, 256, 0, stream>>>(H, Wf2, AGG, nullptr, bf2);
    // logits + log_softmax
    k_final_wmma<<<nb_final, 256, 0, stream>>>(AGG, Wf3, bf3, out);
}